// GraphRestrictedBoltzmannMachine_2654289789161
// MI455X (gfx1250) — compile-verified
//
#include <hip/hip_runtime.h>

// ---------------------------------------------------------------------------
// H(x)[b] = sum_n x[b,n]*h[n] + sum_e x[b,i_e]*x[b,j_e]*J_e
// B=128, N=50000, E=1.6M.
//   1) transpose x (B,N) -> xT (N,128) in d_ws: each edge column becomes a
//      contiguous 512B row; xT (25.6MB) is L2-resident (192MB L2)
//   2) x@h via V_WMMA_F32_16X16X4_F32 (exact fp32), unguarded main loop,
//      2 accumulators for WMMA latency overlap
//   3) edge term: 1 wave = full batch (lane -> 4 batches via float4),
//      2 edges in flight per iteration, 4 atomics per wave at the end
// ---------------------------------------------------------------------------

typedef __attribute__((ext_vector_type(2))) float v2f;
typedef __attribute__((ext_vector_type(8))) float v8f;

#define NB 128  // batch size (fixed by the problem)

__global__ void zero_out_kernel(float* __restrict__ out, int n) {
  int t = blockIdx.x * blockDim.x + threadIdx.x;
  if (t < n) out[t] = 0.0f;
}

// x (B=128 rows, N cols, row-major) -> xT (N rows, 128 cols)
__global__ __launch_bounds__(256)
void transpose_kernel(const float* __restrict__ x, float* __restrict__ xT, int N) {
  __shared__ float tile[32][33];
  const int tx = threadIdx.x;           // 0..31
  const int ty = threadIdx.y;           // 0..7
  const int nb = blockIdx.x;            // N tile
  const int bb = blockIdx.y;            // batch tile (0..3)
#pragma unroll
  for (int r = 0; r < 4; ++r) {
    int b = bb * 32 + ty + r * 8;       // always < 128
    int n = nb * 32 + tx;
    if (n < N) tile[ty + r * 8][tx] = x[(size_t)b * N + n];
  }
  __syncthreads();
#pragma unroll
  for (int r = 0; r < 4; ++r) {
    int n = nb * 32 + ty + r * 8;
    int b = bb * 32 + tx;
    if (n < N) xT[(size_t)n * NB + b] = tile[tx][ty + r * 8];
  }
}

// x@h via V_WMMA_F32_16X16X4_F32.
// A (16x4 f32): lanes 0-15 hold K={0,1} in vgpr0/1; lanes 16-31 hold K={2,3}.
// B (4x16 f32): h chunk replicated across all 16 columns (lane halves give
// K=0/K=2 in vgpr0, K=1/K=3 in vgpr1) -> every D column equals the x@h tile.
__global__ __launch_bounds__(32)
void h_wmma_kernel(const float* __restrict__ x, const float* __restrict__ h,
                   float* __restrict__ out, int N, int kchunk) {
  const int lane  = threadIdx.x & 31;
  const int half  = lane >> 4;          // 0 -> K{0,1}, 1 -> K{2,3}
  const int m     = lane & 15;          // row within 16-row tile
  const int btile = blockIdx.x & 7;     // 128/16 = 8 row tiles
  const int ks    = blockIdx.x >> 3;    // K split id
  const int row   = btile * 16 + m;
  const int kbeg  = ks * kchunk;
  const int kend  = (kbeg + kchunk < N) ? (kbeg + kchunk) : N;
  if (kbeg >= kend) return;

  const float* __restrict__ xrow = x + (size_t)row * N;

  union { v8f v; float f[8]; } acc0, acc1;
  acc0.v = (v8f){0.f, 0.f, 0.f, 0.f, 0.f, 0.f, 0.f, 0.f};
  acc1.v = acc0.v;

  const int len      = kend - kbeg;
  const int kendFull = kbeg + (len & ~7);   // multiple of 8
  const int kendQuad = kbeg + (len & ~3);   // multiple of 4

  int k0 = kbeg;
  // main loop: two independent WMMAs per iteration, no guards
  for (; k0 < kendFull; k0 += 8) {
    const int ka = k0 + half * 2;
    v2f a0, b0, a1, b1;
    a0.x = xrow[ka];     a0.y = xrow[ka + 1];
    b0.x = h[ka];        b0.y = h[ka + 1];
    a1.x = xrow[ka + 4]; a1.y = xrow[ka + 5];
    b1.x = h[ka + 4];    b1.y = h[ka + 5];
    acc0.v = __builtin_amdgcn_wmma_f32_16x16x4_f32(
        false, a0, false, b0, (short)0, acc0.v, false, false);
    acc1.v = __builtin_amdgcn_wmma_f32_16x16x4_f32(
        false, a1, false, b1, (short)0, acc1.v, false, false);
  }
  // 4-wide remainder, still unguarded
  for (; k0 < kendQuad; k0 += 4) {
    const int ka = k0 + half * 2;
    v2f a0, b0;
    a0.x = xrow[ka]; a0.y = xrow[ka + 1];
    b0.x = h[ka];    b0.y = h[ka + 1];
    acc0.v = __builtin_amdgcn_wmma_f32_16x16x4_f32(
        false, a0, false, b0, (short)0, acc0.v, false, false);
  }
  // guarded tail (< 4 elements), at most once
  if (k0 < kend) {
    const int ka = k0 + half * 2;
    const int kb = ka + 1;
    v2f a0, b0;
    a0.x = (ka < kend) ? xrow[ka] : 0.0f;
    a0.y = (kb < kend) ? xrow[kb] : 0.0f;
    b0.x = (ka < kend) ? h[ka] : 0.0f;
    b0.y = (kb < kend) ? h[kb] : 0.0f;
    acc0.v = __builtin_amdgcn_wmma_f32_16x16x4_f32(
        false, a0, false, b0, (short)0, acc0.v, false, false);
  }

  acc0.v = acc0.v + acc1.v;

  // D layout: vgpr r -> M = r (lanes 0-15) / r+8 (lanes 16-31); all N equal.
  if (m == 0) {
    const int base = btile * 16 + half * 8;
#pragma unroll
    for (int r = 0; r < 8; ++r) atomicAdd(&out[base + r], acc0.f[r]);
  }
}

// Edge term: each wave covers the full batch (lane -> float4 of 4 batches),
// grid-strides over edge pairs; 4 b128 gathers in flight per iteration.
__global__ __launch_bounds__(256)
void edge_kernel(const float* __restrict__ xT, const float* __restrict__ J,
                 const int* __restrict__ ei, const int* __restrict__ ej,
                 float* __restrict__ out, int E, int nwaves) {
  const int lane = threadIdx.x & 31;
  const int wid  = blockIdx.x * (blockDim.x >> 5) + (threadIdx.x >> 5);
  const float4* __restrict__ xT4 = reinterpret_cast<const float4*>(xT);

  float4 acc0 = make_float4(0.f, 0.f, 0.f, 0.f);
  float4 acc1 = make_float4(0.f, 0.f, 0.f, 0.f);

  for (int e = wid * 2; e < E; e += 2 * nwaves) {
    const int e1_ok = (e + 1 < E);
    const int   i0  = ei[e];
    const int   j0  = ej[e];
    const float J0  = J[e];
    const int   i1  = e1_ok ? ei[e + 1] : i0;
    const int   j1  = e1_ok ? ej[e + 1] : j0;
    const float J1  = e1_ok ? J[e + 1] : 0.0f;

    const float4 xi0 = xT4[(size_t)i0 * (NB / 4) + lane];
    const float4 xj0 = xT4[(size_t)j0 * (NB / 4) + lane];
    const float4 xi1 = xT4[(size_t)i1 * (NB / 4) + lane];
    const float4 xj1 = xT4[(size_t)j1 * (NB / 4) + lane];

    acc0.x = fmaf(xi0.x * xj0.x, J0, acc0.x);
    acc0.y = fmaf(xi0.y * xj0.y, J0, acc0.y);
    acc0.z = fmaf(xi0.z * xj0.z, J0, acc0.z);
    acc0.w = fmaf(xi0.w * xj0.w, J0, acc0.w);
    acc1.x = fmaf(xi1.x * xj1.x, J1, acc1.x);
    acc1.y = fmaf(xi1.y * xj1.y, J1, acc1.y);
    acc1.z = fmaf(xi1.z * xj1.z, J1, acc1.z);
    acc1.w = fmaf(xi1.w * xj1.w, J1, acc1.w);
  }

  acc0.x += acc1.x; acc0.y += acc1.y; acc0.z += acc1.z; acc0.w += acc1.w;

  atomicAdd(&out[lane * 4 + 0], acc0.x);
  atomicAdd(&out[lane * 4 + 1], acc0.y);
  atomicAdd(&out[lane * 4 + 2], acc0.z);
  atomicAdd(&out[lane * 4 + 3], acc0.w);
}

extern "C" void kernel_launch(void* const* d_in, const int* in_sizes, int n_in,
                              void* d_out, int out_size, void* d_ws, size_t ws_size,
                              hipStream_t stream) {
  const float* x  = (const float*)d_in[0];   // (B, N) f32
  const float* h  = (const float*)d_in[1];   // (N,)   f32
  const float* J  = (const float*)d_in[2];   // (E,)   f32
  const int*   ei = (const int*)d_in[3];     // (E,)   i32
  const int*   ej = (const int*)d_in[4];     // (E,)   i32
  float* out = (float*)d_out;                // (B,)   f32

  const int N = in_sizes[1];
  const int E = in_sizes[2];
  float* xT = (float*)d_ws;                  // N*128 floats (25.6 MB)

  // 1) zero output (harness poisons it)
  zero_out_kernel<<<(out_size + 127) / 128, 128, 0, stream>>>(out, out_size);

  // 2) transpose x -> xT
  dim3 tb(32, 8);
  dim3 tg((N + 31) / 32, NB / 32);
  transpose_kernel<<<tg, tb, 0, stream>>>(x, xT, N);

  // 3) x@h via WMMA (8 row tiles x 64 K splits, 1 wave each)
  const int KSPLIT = 64;
  int kchunk = (((N + KSPLIT - 1) / KSPLIT) + 3) & ~3;
  h_wmma_kernel<<<8 * KSPLIT, 32, 0, stream>>>(x, h, out, N, kchunk);

  // 4) edge gather-reduce: 512 blocks x 8 waves = 4096 waves
  const int blocks = 512, tpb = 256;
  const int nwaves = blocks * (tpb / 32);
  edge_kernel<<<blocks, tpb, 0, stream>>>(xT, J, ei, ej, out, E, nwaves);
}